// GNN_Fingerprinter_49100066128181
// MI455X (gfx1250) — compile-verified
//
#include <hip/hip_runtime.h>
#include <hip/hip_bf16.h>

#define N_NODES 50000
#define N_EDGES 800000
#define IN_CH   128
#define HID_CH  128
#define OUT_CH  64

typedef __attribute__((ext_vector_type(2))) float v2f;
typedef __attribute__((ext_vector_type(8))) float v8f;

// ---------------------------------------------------------------------------
// zero fill
// ---------------------------------------------------------------------------
__global__ void zero_f32(float* __restrict__ p, size_t n) {
    size_t i = (size_t)blockIdx.x * blockDim.x + threadIdx.x;
    size_t stride = (size_t)gridDim.x * blockDim.x;
    for (; i < n; i += stride) p[i] = 0.0f;
}

// ---------------------------------------------------------------------------
// degree: one thread per edge, hardware fp32 atomic into deg[dst]
// ---------------------------------------------------------------------------
__global__ void deg_kernel(const int* __restrict__ dst, float* __restrict__ deg, int E) {
    int e = blockIdx.x * blockDim.x + threadIdx.x;
    if (e < E) unsafeAtomicAdd(&deg[dst[e]], 1.0f);
}

__global__ void invdeg_kernel(const float* __restrict__ deg, float* __restrict__ invdeg, int n) {
    int i = blockIdx.x * blockDim.x + threadIdx.x;
    if (i < n) invdeg[i] = 1.0f / fmaxf(deg[i], 1.0f);
}

// ---------------------------------------------------------------------------
// scatter-add of 128-channel rows: one wave32 per edge, float4 per lane,
// 4 global_atomic_add_f32 per lane (coalesced across the 512B row)
// ---------------------------------------------------------------------------
__global__ __launch_bounds__(256)
void scatter_kernel(const float* __restrict__ feat,
                    const int* __restrict__ src, const int* __restrict__ dst,
                    float* __restrict__ agg, int E) {
    int t    = blockIdx.x * blockDim.x + threadIdx.x;
    int e    = t >> 5;
    int lane = t & 31;
    if (e >= E) return;
    int s = src[e];
    int d = dst[e];
    const float4 v = *(const float4*)(feat + (size_t)s * 128 + lane * 4);
    float* base = agg + (size_t)d * 128 + lane * 4;
    unsafeAtomicAdd(base + 0, v.x);
    unsafeAtomicAdd(base + 1, v.y);
    unsafeAtomicAdd(base + 2, v.z);
    unsafeAtomicAdd(base + 3, v.w);
}

// ---------------------------------------------------------------------------
// Fused SAGE linear:  out = (agg * invdeg) @ Wl + bias + Aself @ Wr  [+ReLU]
// One wave computes a 16x16 output tile via V_WMMA_F32_16X16X4_F32,
// accumulating both GEMM passes into the same fp32 accumulator.
// K is fixed at 128 for both layers.  gridDim.x = 50000/16 = 3125 (exact).
// blockDim.x = 32 * (NOUT/16)  -> each wave owns one 16-wide N tile.
// ---------------------------------------------------------------------------
template <int NOUT, bool RELU>
__global__ __launch_bounds__(2 * NOUT)
void sage_gemm(const float* __restrict__ Aagg,
               const float* __restrict__ Aself,
               const float* __restrict__ invdeg,
               const float* __restrict__ Wl,
               const float* __restrict__ Wr,
               const float* __restrict__ bias,
               float* __restrict__ out) {
    constexpr int K = 128;
    const int wave = threadIdx.x >> 5;     // N tile index within block
    const int lane = threadIdx.x & 31;
    const int row  = lane & 15;            // M within tile (A); N within tile (B, C/D)
    const int hi   = lane >> 4;            // half-wave select
    const int m0   = blockIdx.x * 16;
    const int n0   = wave * 16;

    const float* __restrict__ arow_agg  = Aagg  + (size_t)(m0 + row) * K;
    const float* __restrict__ arow_self = Aself + (size_t)(m0 + row) * K;
    const float  scale = invdeg[m0 + row];

    v8f c = {};

    // pass 1: mean-aggregated neighbors @ Wl   (A scaled on the fly)
    #pragma unroll 4
    for (int k0 = 0; k0 < K; k0 += 4) {
        const int ka = k0 + 2 * hi;
        const float2 t = *(const float2*)(arow_agg + ka);
        v2f a;
        a.x = t.x * scale;
        a.y = t.y * scale;
        v2f b;
        b.x = Wl[(size_t)ka * NOUT + n0 + row];
        b.y = Wl[(size_t)(ka + 1) * NOUT + n0 + row];
        c = __builtin_amdgcn_wmma_f32_16x16x4_f32(false, a, false, b,
                                                  (short)0, c, false, false);
    }

    // pass 2: root features @ Wr  (accumulate into same C)
    #pragma unroll 4
    for (int k0 = 0; k0 < K; k0 += 4) {
        const int ka = k0 + 2 * hi;
        const float2 t = *(const float2*)(arow_self + ka);
        v2f a;
        a.x = t.x;
        a.y = t.y;
        v2f b;
        b.x = Wr[(size_t)ka * NOUT + n0 + row];
        b.y = Wr[(size_t)(ka + 1) * NOUT + n0 + row];
        c = __builtin_amdgcn_wmma_f32_16x16x4_f32(false, a, false, b,
                                                  (short)0, c, false, false);
    }

    // epilogue: bias (+ ReLU), store.  c[r] = D[r + 8*hi][row]
    const float bv = bias[n0 + row];
    #pragma unroll
    for (int r = 0; r < 8; ++r) {
        float v = c[r] + bv;
        if (RELU) v = fmaxf(v, 0.0f);
        out[(size_t)(m0 + r + 8 * hi) * NOUT + n0 + row] = v;
    }
}

// ---------------------------------------------------------------------------
// launch
// ---------------------------------------------------------------------------
extern "C" void kernel_launch(void* const* d_in, const int* in_sizes, int n_in,
                              void* d_out, int out_size, void* d_ws, size_t ws_size,
                              hipStream_t stream) {
    const float* x   = (const float*)d_in[0];
    const int*   ei  = (const int*)d_in[1];     // [2, E]
    const float* W1l = (const float*)d_in[2];
    const float* b1  = (const float*)d_in[3];
    const float* W1r = (const float*)d_in[4];
    const float* W2l = (const float*)d_in[5];
    const float* b2  = (const float*)d_in[6];
    const float* W2r = (const float*)d_in[7];

    const int* src = ei;
    const int* dst = ei + N_EDGES;

    // workspace layout (floats)
    float* agg    = (float*)d_ws;                          // [N, 128]
    float* h      = agg + (size_t)N_NODES * 128;           // [N, 128]
    float* deg    = h + (size_t)N_NODES * 128;             // [N]
    float* invdeg = deg + N_NODES;                         // [N]

    const int ZB = 256;

    // degree (once, shared by both layers)
    zero_f32<<<1024, ZB, 0, stream>>>(deg, (size_t)N_NODES);
    deg_kernel<<<(N_EDGES + ZB - 1) / ZB, ZB, 0, stream>>>(dst, deg, N_EDGES);
    invdeg_kernel<<<(N_NODES + ZB - 1) / ZB, ZB, 0, stream>>>(deg, invdeg, N_NODES);

    // ---- layer 1 ----
    zero_f32<<<2048, ZB, 0, stream>>>(agg, (size_t)N_NODES * 128);
    scatter_kernel<<<(N_EDGES * 32 + ZB - 1) / ZB, ZB, 0, stream>>>(x, src, dst, agg, N_EDGES);
    sage_gemm<HID_CH, true><<<N_NODES / 16, 2 * HID_CH, 0, stream>>>(
        agg, x, invdeg, W1l, W1r, b1, h);

    // ---- layer 2 ----
    zero_f32<<<2048, ZB, 0, stream>>>(agg, (size_t)N_NODES * 128);
    scatter_kernel<<<(N_EDGES * 32 + ZB - 1) / ZB, ZB, 0, stream>>>(h, src, dst, agg, N_EDGES);
    sage_gemm<OUT_CH, false><<<N_NODES / 16, 2 * OUT_CH, 0, stream>>>(
        agg, h, invdeg, W2l, W2r, b2, (float*)d_out);
}